// Net_49512382988901
// MI455X (gfx1250) — compile-verified
//
#include <hip/hip_runtime.h>
#include <math.h>

// ---------------------------------------------------------------------------
// Problem constants (match reference)
// ---------------------------------------------------------------------------
#define PP   30      // points per cloud
#define KNN  20      // neighbors
#define KSP  5       // spline control points
#define FN   10      // filters
#define EPSF 1e-8f

typedef __attribute__((ext_vector_type(2))) float v2f;
typedef __attribute__((ext_vector_type(8))) float v8f;

static __device__ __forceinline__ v8f wmma_f32_4(v2f a, v2f b, v8f c) {
    // D = A(16x4 f32) * B(4x16 f32) + C(16x16 f32)
    return __builtin_amdgcn_wmma_f32_16x16x4_f32(
        /*neg_a=*/false, a, /*neg_b=*/false, b,
        /*c_mod=*/(short)0, c, /*reuse_a=*/false, /*reuse_b=*/false);
}

// ---------------------------------------------------------------------------
// Kernel 1: per-cloud directional spline conv features -> ys[B][12] (padded)
// One wave32 per cloud; lane i handles point i (i < PP).
// ---------------------------------------------------------------------------
__global__ __launch_bounds__(256) void dsc_kernel(
    const float* __restrict__ pos,   // [B*PP, 3]
    const float* __restrict__ W,     // [KSP, FN]
    const float* __restrict__ bdsc,  // [FN]
    float* __restrict__ ys)          // [B, 12] (cols 10,11 zero)
{
    __shared__ float spx[8][32], spy[8][32], spz[8][32];
    __shared__ float sW[KSP][FN];
    __shared__ float sbd[FN];

    const int tid  = threadIdx.x;
    const int wave = tid >> 5;
    const int lane = tid & 31;
    const int cloud = blockIdx.x * 8 + wave;

    float mx = 0.f, my = 0.f, mz = 0.f;
    if (lane < PP) {
        const float* pp = pos + ((size_t)cloud * PP + lane) * 3;
        mx = pp[0]; my = pp[1]; mz = pp[2];
    }
    spx[wave][lane] = mx; spy[wave][lane] = my; spz[wave][lane] = mz;
    if (tid < KSP * FN) sW[tid / FN][tid % FN] = W[tid];
    if (tid < FN)       sbd[tid] = bdsc[tid];
    __syncthreads();

    // pairwise squared distances to all candidates; +1e10 on self (as ref)
    float d2[PP];
#pragma unroll
    for (int c = 0; c < PP; ++c) {
        float dx = spx[wave][c] - mx;
        float dy = spy[wave][c] - my;
        float dz = spz[wave][c] - mz;
        float d  = dx * dx + dy * dy + dz * dz;
        d2[c] = (c == lane) ? d + 1e10f : d;
    }

    // rank-based KNN selection (order-invariant set == top_k(-d2) set,
    // ties broken by lower index first, matching jax.lax.top_k)
    int rank[PP];
#pragma unroll
    for (int c = 0; c < PP; ++c) rank[c] = 0;
#pragma unroll
    for (int c = 0; c < PP; ++c) {
#pragma unroll
        for (int cp = c + 1; cp < PP; ++cp) {
            bool t = d2[cp] < d2[c];
            rank[c]  += t ? 1 : 0;
            rank[cp] += t ? 0 : 1;
        }
    }
    unsigned member = 0u;
#pragma unroll
    for (int c = 0; c < PP; ++c)
        if (rank[c] < KNN) member |= (1u << c);

    // scale = max_{members} sqrt(d2+EPS) + EPS
    float scale = 0.f;
#pragma unroll
    for (int c = 0; c < PP; ++c) {
        float nm = sqrtf(d2[c] + EPSF);
        scale = ((member >> c) & 1u) ? fmaxf(scale, nm) : scale;
    }
    scale += EPSF;
    const float inv = 1.0f / scale;

    // 3x3 covariance of normalized offsets over members
    float cxx = 0.f, cxy = 0.f, cxz = 0.f, cyy = 0.f, cyz = 0.f, czz = 0.f;
#pragma unroll
    for (int c = 0; c < PP; ++c) {
        float rx = (spx[wave][c] - mx) * inv;
        float ry = (spy[wave][c] - my) * inv;
        float rz = (spz[wave][c] - mz) * inv;
        float m  = ((member >> c) & 1u) ? 1.f : 0.f;
        cxx += m * rx * rx; cxy += m * rx * ry; cxz += m * rx * rz;
        cyy += m * ry * ry; cyz += m * ry * rz; czz += m * rz * rz;
    }

    // power iteration, 8 steps
    float vx = 0.57735027f, vy = 0.57735027f, vz = 0.57735027f;
#pragma unroll
    for (int it = 0; it < 8; ++it) {
        float wx = cxx * vx + cxy * vy + cxz * vz;
        float wy = cxy * vx + cyy * vy + cyz * vz;
        float wz = cxz * vx + cyz * vy + czz * vz;
        float r  = 1.0f / (sqrtf(wx * wx + wy * wy + wz * wz) + EPSF);
        vx = wx * r; vy = wy * r; vz = wz * r;
    }

    // spline weights accumulated per control point (avoids dynamic W index)
    float wacc[KSP];
#pragma unroll
    for (int r = 0; r < KSP; ++r) wacc[r] = 0.f;
#pragma unroll
    for (int c = 0; c < PP; ++c) {
        float rx = (spx[wave][c] - mx) * inv;
        float ry = (spy[wave][c] - my) * inv;
        float rz = (spz[wave][c] - mz) * inv;
        float t  = rx * vx + ry * vy + rz * vz;
        float u  = fminf(fmaxf((t + 1.f) * 0.5f, 0.f), 1.f) * (float)(KSP - 1);
        float fi = fminf(floorf(u), (float)(KSP - 2));
        float fr = u - fi;
        int   i0 = (int)fi;
        float m  = ((member >> c) & 1u) ? 1.f : 0.f;
#pragma unroll
        for (int r = 0; r < KSP; ++r) {
            float w = (r == i0) ? (1.f - fr) : ((r == i0 + 1) ? fr : 0.f);
            wacc[r] += m * w;
        }
    }

    // y = (sum_r wacc[r]*W[r,:]) / K + b; sigmoid; mean over points in cloud
    float sacc[FN];
#pragma unroll
    for (int f = 0; f < FN; ++f) {
        float y = 0.f;
#pragma unroll
        for (int r = 0; r < KSP; ++r) y += wacc[r] * sW[r][f];
        y = y * (1.0f / KNN) + sbd[f];
        float s = 1.0f / (1.0f + expf(-y));
        s = (lane < PP) ? s : 0.f;
#pragma unroll
        for (int d = 1; d < 32; d <<= 1) s += __shfl_xor(s, d, 32);
        sacc[f] = s * (1.0f / PP);
    }

    // lane f writes column f (cols 10,11 -> 0 padding for WMMA K=12)
    float outv = 0.f;
#pragma unroll
    for (int f = 0; f < FN; ++f) outv = (lane == f) ? sacc[f] : outv;
    if (lane < 12) ys[(size_t)cloud * 12 + lane] = outv;
}

// ---------------------------------------------------------------------------
// Kernel 2: fused MLP  h=elu(ys@w1+b1); logits=h@w2+b2; log_softmax
// 4 waves per block; each wave owns a 16-row tile, WMMA f32 16x16x4.
// Dynamic LDS layout (floats):
//   w1t [256][12]          (w1 transposed, K padded 10->12)
//   w2t [48][260]          (w2 transposed, N padded 40->48, stride 260)
//   b2s [48], b1s [256]
//   hbuf[4][16][260]       (per-wave h tile, stride 260)
// ---------------------------------------------------------------------------
__global__ __launch_bounds__(128) void mlp_kernel(
    const float* __restrict__ ys,   // [B,12]
    const float* __restrict__ w1,   // [10,256]
    const float* __restrict__ b1,   // [256]
    const float* __restrict__ w2,   // [256,40]
    const float* __restrict__ b2,   // [40]
    float* __restrict__ out)        // [B,40] log-probs
{
    extern __shared__ float smem[];
    float* w1t  = smem;                    // 256*12 = 3072
    float* w2t  = w1t + 256 * 12;          // 48*260 = 12480
    float* b2s  = w2t + 48 * 260;          // 48
    float* b1s  = b2s + 48;                // 256
    float* hbuf = b1s + 256;               // 4*16*260 = 16640

    const int tid = threadIdx.x;

    // cooperative staging
    for (int i = tid; i < 256 * 12; i += blockDim.x) {
        int n = i / 12, k = i % 12;
        w1t[i] = (k < FN) ? w1[k * 256 + n] : 0.f;
    }
    for (int i = tid; i < 48 * 256; i += blockDim.x) {
        int n = i / 256, k = i % 256;
        w2t[n * 260 + k] = (n < 40) ? w2[k * 40 + n] : 0.f;
    }
    for (int i = tid; i < 48;  i += blockDim.x) b2s[i] = (i < 40) ? b2[i] : 0.f;
    for (int i = tid; i < 256; i += blockDim.x) b1s[i] = b1[i];
    __syncthreads();

    const int wave = tid >> 5;
    const int lane = tid & 31;
    const int lh   = lane & 15;      // position within 16-lane half
    const int half = lane >> 4;      // 0 or 1
    const int row0 = (blockIdx.x * 4 + wave) * 16;   // grid sized so always valid
    float* hw = hbuf + wave * 16 * 260;

    // ---- GEMM1: h[16][256] = elu(ys_tile[16][12] @ w1t + b1) ----
    // A fragment: lane -> (M=lh, K=4*kk + 2*half + {0,1}); even element
    // offsets keep v2f loads 8-byte aligned.
    const float* arow = ys + (size_t)(row0 + lh) * 12 + 2 * half;
    v2f afrag[3];
#pragma unroll
    for (int kk = 0; kk < 3; ++kk)
        afrag[kk] = *(const v2f*)(arow + 4 * kk);

    for (int nt = 0; nt < 16; ++nt) {
        const int n0 = nt * 16;
        v8f acc = {};
#pragma unroll
        for (int kk = 0; kk < 3; ++kk) {
            v2f bfrag = *(const v2f*)(w1t + (n0 + lh) * 12 + 4 * kk + 2 * half);
            acc = wmma_f32_4(afrag[kk], bfrag, acc);
        }
        const float bias = b1s[n0 + lh];
#pragma unroll
        for (int v = 0; v < 8; ++v) {
            float x = acc[v] + bias;                 // D: M=v+8*half, N=lh
            x = (x > 0.f) ? x : (expf(x) - 1.0f);    // ELU (alpha=1)
            hw[(v + 8 * half) * 260 + n0 + lh] = x;
        }
    }
    __syncthreads();   // all waves full -> uniform barrier

    // ---- GEMM2: logits[16][48] = h[16][256] @ w2t ----
    v8f acc2[3] = {v8f{}, v8f{}, v8f{}};
    for (int k4 = 0; k4 < 64; ++k4) {
        v2f a = *(const v2f*)(hw + lh * 260 + 4 * k4 + 2 * half);
#pragma unroll
        for (int nt = 0; nt < 3; ++nt) {
            v2f b = *(const v2f*)(w2t + (nt * 16 + lh) * 260 + 4 * k4 + 2 * half);
            acc2[nt] = wmma_f32_4(a, b, acc2[nt]);
        }
    }

    // ---- bias + log_softmax over the 40 valid columns of each row ----
    // Row (v, half) lives entirely inside one 16-lane half: cols nt*16+lh.
    const float bia0 = b2s[lh];
    const float bia1 = b2s[16 + lh];
    const float bia2 = b2s[32 + lh];
    const bool  valid2 = (lh < 8);   // cols 32..39 only
#pragma unroll
    for (int v = 0; v < 8; ++v) {
        float x0 = acc2[0][v] + bia0;
        float x1 = acc2[1][v] + bia1;
        float x2 = valid2 ? (acc2[2][v] + bia2) : -3.0e38f;
        float m = fmaxf(fmaxf(x0, x1), x2);
#pragma unroll
        for (int d = 1; d < 16; d <<= 1) m = fmaxf(m, __shfl_xor(m, d, 32));
        float s = expf(x0 - m) + expf(x1 - m) + (valid2 ? expf(x2 - m) : 0.f);
#pragma unroll
        for (int d = 1; d < 16; d <<= 1) s += __shfl_xor(s, d, 32);
        const float ls = m + logf(s);
        float* orow = out + (size_t)(row0 + v + 8 * half) * 40;
        orow[lh]      = x0 - ls;
        orow[16 + lh] = x1 - ls;
        if (valid2) orow[32 + lh] = x2 - ls;
    }
}

// ---------------------------------------------------------------------------
// Host launcher
// ---------------------------------------------------------------------------
extern "C" void kernel_launch(void* const* d_in, const int* in_sizes, int n_in,
                              void* d_out, int out_size, void* d_ws, size_t ws_size,
                              hipStream_t stream) {
    const float* pos  = (const float*)d_in[0];
    const float* W    = (const float*)d_in[1];
    const float* bdsc = (const float*)d_in[2];
    const float* w1   = (const float*)d_in[3];
    const float* b1   = (const float*)d_in[4];
    const float* w2   = (const float*)d_in[5];
    const float* b2   = (const float*)d_in[6];
    float* out = (float*)d_out;

    const int npts    = in_sizes[0] / 3;   // B*P
    const int nclouds = npts / PP;         // 16384

    float* ys = (float*)d_ws;              // [nclouds][12] scratch

    dsc_kernel<<<nclouds / 8, 256, 0, stream>>>(pos, W, bdsc, ys);

    const size_t smem_bytes =
        (size_t)(256 * 12 + 48 * 260 + 48 + 256 + 4 * 16 * 260) * sizeof(float);
    mlp_kernel<<<nclouds / 64, 128, smem_bytes, stream>>>(ys, w1, b1, w2, b2, out);
}